// HairBundleSDE_23965917512008
// MI455X (gfx1250) — compile-verified
//
#include <hip/hip_runtime.h>

// ---------------------------------------------------------------------------
// HairBundle SDE derivative: pure streaming elementwise kernel.
// Memory-bound (AI ~ 0.75 flop/byte); target = saturate 23.3 TB/s HBM with
// b128 non-temporal loads/stores. No matrix structure -> no WMMA by design.
// ---------------------------------------------------------------------------

typedef __attribute__((ext_vector_type(4))) float v4f;

// Model constants (folded: divisions -> reciprocal multiplies)
#define HB_D0        0.5f
#define HB_DELTA_INV 5.0f          // 1/0.2
#define HB_K_GS      0.75f
#define HB_K_SP      0.6f
#define HB_LAM_A_INV 0.1f          // 1/10
#define HB_S_CA      0.7f
#define HB_C_M       1.5f
#define HB_C_GS      1.2f
#define HB_C_T       0.8f
#define HB_TAUGS_INV 0.2f          // 1/5
#define HB_TAUT_INV  0.1f          // 1/10
#define HB_AMP       0.3f
#define HB_OMEGA     6.283185307179586f

__device__ __forceinline__ float sigmoid_fast(float z) {
  // 1/(1+exp(-z)) via v_exp_f32 + v_rcp_f32
  float e = __expf(-z);
  return __builtin_amdgcn_rcpf(1.0f + e);
}

__device__ __forceinline__ void hb_deriv(float X, float Xa, float pm, float pgs, float pt,
                                         float force,
                                         float& dX, float& dXa, float& dpm,
                                         float& dpgs, float& dpt) {
  const float z      = (X - Xa - HB_D0) * HB_DELTA_INV;
  const float p_open = sigmoid_fast(z);
  const float f_gs   = HB_K_GS * (X - Xa - HB_D0 * p_open);
  dX   = -f_gs - HB_K_SP * X + force;                       // LAM = 1
  dXa  = (f_gs - (1.0f - HB_S_CA * pm)) * HB_LAM_A_INV;     // F_MAX = 1
  dpm  = HB_C_M * p_open * (1.0f - pm) - pm;                // TAU_M = 1
  dpgs = (HB_C_GS * p_open * (1.0f - pgs) - pgs) * HB_TAUGS_INV;
  dpt  = (HB_C_T  * p_open * (1.0f - pt ) - pt ) * HB_TAUT_INV;
}

__global__ __launch_bounds__(256) void HairBundleSDE_kernel(
    const float* __restrict__ t,
    const float* __restrict__ x,
    float* __restrict__ out,
    int n) {
  // Uniform forcing term (scalar load + v_sin_f32, hoisted once per thread)
  const float force = HB_AMP * __sinf(HB_OMEGA * t[0]);

  const long long gid = (long long)blockIdx.x * blockDim.x + threadIdx.x;
  const long long i0  = gid * 4;          // 4 samples / thread = 80B = 5 x b128

  if (i0 + 4 <= (long long)n) {
    const v4f* pv = (const v4f*)(x + i0 * 5);   // 16B-aligned: 80B per-thread stride
    const v4f a = __builtin_nontemporal_load(pv + 0);
    const v4f b = __builtin_nontemporal_load(pv + 1);
    const v4f c = __builtin_nontemporal_load(pv + 2);
    const v4f d = __builtin_nontemporal_load(pv + 3);
    const v4f e = __builtin_nontemporal_load(pv + 4);

    float r[20];
    // AoS layout: [X,Xa,pm,pgs,pt] x 4 samples packed across a..e
    hb_deriv(a.x, a.y, a.z, a.w, b.x, force, r[0],  r[1],  r[2],  r[3],  r[4]);
    hb_deriv(b.y, b.z, b.w, c.x, c.y, force, r[5],  r[6],  r[7],  r[8],  r[9]);
    hb_deriv(c.z, c.w, d.x, d.y, d.z, force, r[10], r[11], r[12], r[13], r[14]);
    hb_deriv(d.w, e.x, e.y, e.z, e.w, force, r[15], r[16], r[17], r[18], r[19]);

    const v4f o0 = {r[0],  r[1],  r[2],  r[3]};
    const v4f o1 = {r[4],  r[5],  r[6],  r[7]};
    const v4f o2 = {r[8],  r[9],  r[10], r[11]};
    const v4f o3 = {r[12], r[13], r[14], r[15]};
    const v4f o4 = {r[16], r[17], r[18], r[19]};

    v4f* ov = (v4f*)(out + i0 * 5);
    __builtin_nontemporal_store(o0, ov + 0);
    __builtin_nontemporal_store(o1, ov + 1);
    __builtin_nontemporal_store(o2, ov + 2);
    __builtin_nontemporal_store(o3, ov + 3);
    __builtin_nontemporal_store(o4, ov + 4);
  } else if (i0 < (long long)n) {
    // Scalar tail (n % 4 != 0); dword loads are always aligned.
    for (long long i = i0; i < (long long)n; ++i) {
      const float* p = x + i * 5;
      float dX, dXa, dpm, dpgs, dpt;
      hb_deriv(p[0], p[1], p[2], p[3], p[4], force, dX, dXa, dpm, dpgs, dpt);
      float* o = out + i * 5;
      o[0] = dX; o[1] = dXa; o[2] = dpm; o[3] = dpgs; o[4] = dpt;
    }
  }
}

extern "C" void kernel_launch(void* const* d_in, const int* in_sizes, int n_in,
                              void* d_out, int out_size, void* d_ws, size_t ws_size,
                              hipStream_t stream) {
  const float* t  = (const float*)d_in[0];   // shape (1,)
  const float* x  = (const float*)d_in[1];   // shape (N, 5) row-major f32
  float* out      = (float*)d_out;           // shape (N, 5) f32

  const int n       = in_sizes[1] / 5;       // number of samples
  const int nvec    = (n + 3) / 4;           // 4 samples per thread
  const int threads = 256;                   // 8 wave32 waves per block
  const int blocks  = (nvec + threads - 1) / threads;

  HairBundleSDE_kernel<<<blocks, threads, 0, stream>>>(t, x, out, n);
}